// PatchPooling_32469952758228
// MI455X (gfx1250) — compile-verified
//
#include <hip/hip_runtime.h>

typedef __attribute__((ext_vector_type(2))) float v2f;
typedef __attribute__((ext_vector_type(8))) float v8f;

#define B_ 64
#define S_ 1024
#define D_ 256
#define P_ 64

__device__ __forceinline__ int imin(int a, int b) { return a < b ? a : b; }
__device__ __forceinline__ int imax(int a, int b) { return a > b ? a : b; }

__global__ __launch_bounds__(256) void PatchPooling_wmma_kernel(
    const float* __restrict__ batch,
    const int*   __restrict__ plen,
    float*       __restrict__ out)
{
    // Exclusive-scan patch boundaries for this block's batch, in LDS.
    __shared__ int s_bound[P_ + 1];

    const int b     = blockIdx.x >> 2;   // batch index (block-uniform)
    const int ptile = blockIdx.x & 3;    // patch tile of 16
    const int tid   = threadIdx.x;
    const int wave  = tid >> 5;          // 0..7 -> owns channel cols [32w, 32w+32)
    const int lane  = tid & 31;
    const int m     = lane & 15;         // A row (patch) AND B/C/D column-in-tile
    const int khalf = lane >> 4;         // lane-half selects K offset per ISA layout

    if (tid == 0) s_bound[0] = 0;
    if (tid < P_) s_bound[tid + 1] = plen[b * P_ + tid];
    __syncthreads();
    if (tid == 0) {
        int acc = 0;
        for (int p = 0; p < P_; ++p) { acc += s_bound[p + 1]; s_bound[p + 1] = acc; }
    }
    __syncthreads();

    const int p_base = ptile * 16;

    // This lane's A-row segment boundaries, clamped to [0,S] so the 0/1 mask
    // also kills any (clamped) row at s >= S. Branchless everywhere.
    const int fr = imin(imax(s_bound[p_base + m], 0), S_);
    const int to = imin(imax(s_bound[p_base + m + 1], 0), S_);

    // The 16 patches of this tile cover a contiguous K range; iterate only that.
    const int k_lo = s_bound[p_base] & ~3;
    const int k_hi = s_bound[p_base + 16];
    const int nsteps = (k_hi > k_lo) ? ((k_hi - k_lo + 3) >> 2) : 0;

    const int col0 = wave * 32 + m;
    const int col1 = col0 + 16;
    // Block-uniform base: loads become SADDR(base SGPR pair) + 32-bit VGPR offset.
    const float* __restrict__ bbase = batch + (size_t)b * S_ * D_;

    v8f acc0 = {};
    v8f acc1 = {};

#pragma unroll 2
    for (int i = 0; i < nsteps; ++i) {
        const int k  = k_lo + 4 * i;
        // A 16x4 f32 layout: lanes 0-15 hold K=k+0 (VGPR0), k+1 (VGPR1);
        // lanes 16-31 hold K=k+2, k+3.
        const int s0 = k + 2 * khalf;
        const int s1 = s0 + 1;
        v2f a;
        a.x = (s0 >= fr && s0 < to) ? 1.0f : 0.0f;
        a.y = (s1 >= fr && s1 < to) ? 1.0f : 0.0f;
        // Clamp row index instead of predicating the load: stays in-bounds,
        // stays convergent (EXEC all-1s for WMMA), garbage rows get mask 0.
        // 32-bit offsets (max 1024*256 floats) keep address math to one
        // v_lshl_add per load instead of 64-bit VGPR-pair adds.
        const int o0 = imin(s0, S_ - 1) << 8;   // r * D_ (D_ = 256)
        const int o1 = imin(s1, S_ - 1) << 8;
        // B 4x16 f32: lane = column n, lane-half/VGPR select the K row.
        v2f bv0, bv1;
        bv0.x = bbase[o0 + col0];
        bv0.y = bbase[o1 + col0];
        bv1.x = bbase[o0 + col1];
        bv1.y = bbase[o1 + col1];
        acc0 = __builtin_amdgcn_wmma_f32_16x16x4_f32(
            false, a, false, bv0, (short)0, acc0, false, false);
        acc1 = __builtin_amdgcn_wmma_f32_16x16x4_f32(
            false, a, false, bv1, (short)0, acc1, false, false);
    }

    // C/D 16x16 f32 layout: lane holds rows M = v + 8*khalf in VGPR v, column = m.
    float* __restrict__ optr = out + (size_t)b * P_ * D_;
#pragma unroll
    for (int v = 0; v < 8; ++v) {
        const int p   = p_base + v + 8 * khalf;
        const int frc = imin(imax(s_bound[p], 0), S_);
        const int toc = imin(imax(s_bound[p + 1], 0), S_);
        const int cnt = toc - frc;
        const float fden = (float)imax(cnt, 1);
        // Unconditional IEEE f32 divide (matches reference seg_sum/denom),
        // then branchless select of -1 for empty patches.
        const float q0 = acc0[v] / fden;
        const float q1 = acc1[v] / fden;
        const int   po = p << 8;              // p * D_, 32-bit store offsets
        optr[po + col0] = (cnt == 0) ? -1.0f : q0;
        optr[po + col1] = (cnt == 0) ? -1.0f : q1;
    }
}

extern "C" void kernel_launch(void* const* d_in, const int* in_sizes, int n_in,
                              void* d_out, int out_size, void* d_ws, size_t ws_size,
                              hipStream_t stream) {
    (void)in_sizes; (void)n_in; (void)d_ws; (void)ws_size; (void)out_size;
    const float* batch = (const float*)d_in[0];
    const int*   plen  = (const int*)d_in[1];
    float*       out   = (float*)d_out;

    // One block per (batch, patch-tile): 8 waves x 2 channel-tiles = all 256 cols.
    dim3 grid(B_ * (P_ / 16));   // 256 blocks
    dim3 block(256);
    PatchPooling_wmma_kernel<<<grid, block, 0, stream>>>(batch, plen, out);
}